// checkerboard_13924283973960
// MI455X (gfx1250) — compile-verified
//
#include <hip/hip_runtime.h>
#include <hip/hip_bf16.h>

typedef __attribute__((ext_vector_type(2))) float v2f;
typedef __attribute__((ext_vector_type(8))) float v8f;

#define HAVE_WMMA_F32_4 __has_builtin(__builtin_amdgcn_wmma_f32_16x16x4_f32)

struct GateArgs { int n; int hi[6]; int lo[6]; int gid[6]; };
struct AddrArgs { unsigned lm; int tb; int ls; int hs; };

__device__ __forceinline__ float2 cmul(float2 a, float2 b) {
    return make_float2(a.x * b.x - a.y * b.y, a.x * b.y + a.y * b.x);
}

__device__ __forceinline__ void mm4(float2 C[4][4], const float2 A[4][4], const float2 B[4][4]) {
    for (int r = 0; r < 4; ++r)
        for (int c = 0; c < 4; ++c) {
            float2 acc = make_float2(0.f, 0.f);
            for (int k = 0; k < 4; ++k) {
                float2 p = cmul(A[r][k], B[k][c]);
                acc.x += p.x; acc.y += p.y;
            }
            C[r][c] = acc;
        }
}

// One thread per gate: U = expm(H - H^dagger), emit 16x16 real G = diag(W, W),
// W = [[Ur, -Ui], [Ui, Ur]]  (row-major 16x16 floats per gate in workspace).
__global__ void expm_g_kernel(const float2* __restrict__ H1,
                              const float2* __restrict__ H2,
                              float* __restrict__ Gout) {
    int g = blockIdx.x * blockDim.x + threadIdx.x;
    if (g >= 120) return;
    int l = g / 20, rem = g % 20, blk = rem / 10, i = rem % 10;
    const float2* H = (blk == 0 ? H1 : H2) + (size_t)(l * 10 + i) * 16;

    float2 A[4][4];
    for (int r = 0; r < 4; ++r)
        for (int c = 0; c < 4; ++c) {
            float2 h = H[r * 4 + c], hd = H[c * 4 + r];
            A[r][c] = make_float2(h.x - hd.x, h.y + hd.y);  // H - conj(H^T)
        }

    // scaling: s such that ||A||/2^s <= 0.5 (max abs row sum bound)
    float nrm = 0.f;
    for (int r = 0; r < 4; ++r) {
        float rs = 0.f;
        for (int c = 0; c < 4; ++c) rs += fabsf(A[r][c].x) + fabsf(A[r][c].y);
        nrm = fmaxf(nrm, rs);
    }
    int s = 0; float t = nrm;
    while (t > 0.5f && s < 30) { t *= 0.5f; ++s; }
    float sc = ldexpf(1.0f, -s);
    for (int r = 0; r < 4; ++r)
        for (int c = 0; c < 4; ++c) { A[r][c].x *= sc; A[r][c].y *= sc; }

    // order-12 Taylor, Horner: T = I + A(I + A/2(I + ... (I + A/12)))
    float2 T[4][4], M[4][4];
    for (int r = 0; r < 4; ++r)
        for (int c = 0; c < 4; ++c) T[r][c] = make_float2(r == c ? 1.f : 0.f, 0.f);
    for (int k = 12; k >= 1; --k) {
        mm4(M, A, T);
        float inv = 1.f / (float)k;
        for (int r = 0; r < 4; ++r)
            for (int c = 0; c < 4; ++c)
                T[r][c] = make_float2((r == c ? 1.f : 0.f) + M[r][c].x * inv,
                                      M[r][c].y * inv);
    }
    for (int q = 0; q < s; ++q) {  // square back
        mm4(M, T, T);
        for (int r = 0; r < 4; ++r)
            for (int c = 0; c < 4; ++c) T[r][c] = M[r][c];
    }

    float* out = Gout + (size_t)g * 256;
    for (int r = 0; r < 16; ++r)
        for (int c = 0; c < 16; ++c) {
            float val = 0.f;
            if ((r >> 3) == (c >> 3)) {
                int rr = r & 7, cc = c & 7, ri = rr & 3, ci = cc & 3;
                float2 u = T[ri][ci];
                val = (rr < 4) ? (cc < 4 ? u.x : -u.y) : (cc < 4 ? u.y : u.x);
            }
            out[r * 16 + c] = val;
        }
}

__device__ __forceinline__ unsigned ins2(unsigned x, int b1, int b2) {
    // insert zero bit at b1 then b2 (b1 < b2, final coordinates)
    x = ((x >> b1) << (b1 + 1)) | (x & ((1u << b1) - 1u));
    x = ((x >> b2) << (b2 + 1)) | (x & ((1u << b2) - 1u));
    return x;
}

// Apply up to 6 disjoint 2-bit gates to a 2^13-amplitude LDS tile.
// Tile -> global map: glob = (local & lm) | (tile << tb) | ((local >> ls) << hs)
__global__ __launch_bounds__(256) void apply_block_kernel(float2* __restrict__ psi,
                                                          const float* __restrict__ G,
                                                          AddrArgs aa, GateArgs ga) {
    __shared__ __align__(16) float2 tile[8192];
    const int tid = threadIdx.x;
    const unsigned tileid = blockIdx.x;

    for (int it = 0; it < 16; ++it) {
        unsigned local = ((unsigned)(tid + it * 256)) * 2u;
        unsigned glob = (local & aa.lm) | (tileid << aa.tb) | ((local >> aa.ls) << aa.hs);
        *(float4*)&tile[local] = *(const float4*)&psi[glob];
    }
    __syncthreads();

    const int lane = tid & 31;
    const int wave = tid >> 5;
    const int n = lane & 15;   // wmma column
    const int h = lane >> 4;   // lane half

    for (int gi = 0; gi < ga.n; ++gi) {
        const int hi = ga.hi[gi], lo = ga.lo[gi];
        const int bl = hi < lo ? hi : lo, bh = hi < lo ? lo : hi;
        const unsigned oh = 1u << hi, ol = 1u << lo;
        const float* Gm = G + (size_t)ga.gid[gi] * 256;
#if HAVE_WMMA_F32_4
        // A fragments: A[m][K], m = lane%16, K = 4c + 2h + {0,1}
        const float* Gr = Gm + n * 16;
        const int kb = 2 * h;
        v2f a0 = {Gr[0 + kb],  Gr[1 + kb]};
        v2f a1 = {Gr[4 + kb],  Gr[5 + kb]};
        v2f a2 = {Gr[8 + kb],  Gr[9 + kb]};
        v2f a3 = {Gr[12 + kb], Gr[13 + kb]};
#endif
        for (int q = 0; q < 8; ++q) {
            unsigned gA = (unsigned)((wave * 8 + q) * 32 + n);
            unsigned gB = gA + 16u;
            unsigned tA = ins2(gA, bl, bh), tB = ins2(gB, bl, bh);
#if HAVE_WMMA_F32_4
            unsigned hb = ((unsigned)h) << hi;
            float2 A0 = tile[tA + hb], A1 = tile[tA + hb + ol];
            float2 B0 = tile[tB + hb], B1 = tile[tB + hb + ol];
            // B fragments: In[K][n], K rows 0-3 Re grpA, 4-7 Im grpA, 8-11 Re grpB, 12-15 Im grpB
            v2f b0 = {A0.x, A1.x}, b1 = {A0.y, A1.y};
            v2f b2 = {B0.x, B1.x}, b3 = {B0.y, B1.y};
            v8f d = {0.f, 0.f, 0.f, 0.f, 0.f, 0.f, 0.f, 0.f};
            d = __builtin_amdgcn_wmma_f32_16x16x4_f32(false, a0, false, b0, (short)0, d, false, false);
            d = __builtin_amdgcn_wmma_f32_16x16x4_f32(false, a1, false, b1, (short)0, d, false, false);
            d = __builtin_amdgcn_wmma_f32_16x16x4_f32(false, a2, false, b2, (short)0, d, false, false);
            d = __builtin_amdgcn_wmma_f32_16x16x4_f32(false, a3, false, b3, (short)0, d, false, false);
            // D rows M = v + 8h: lane half selects group; amps 0..3 Re in d[0..3], Im in d[4..7]
            unsigned tO = h ? tB : tA;
            tile[tO]           = make_float2(d[0], d[4]);
            tile[tO + ol]      = make_float2(d[1], d[5]);
            tile[tO + oh]      = make_float2(d[2], d[6]);
            tile[tO + oh + ol] = make_float2(d[3], d[7]);
#else
            // scalar fallback: each lane applies the 8x8 real W to its own group
            unsigned tO = h ? tB : tA;
            float2 v0 = tile[tO], v1 = tile[tO + ol], v2 = tile[tO + oh], v3 = tile[tO + oh + ol];
            float vr[8] = {v0.x, v1.x, v2.x, v3.x, v0.y, v1.y, v2.y, v3.y};
            float2 o[4];
            for (int k = 0; k < 4; ++k) {
                float re = 0.f, im = 0.f;
                for (int j = 0; j < 8; ++j) {
                    re += Gm[k * 16 + j] * vr[j];
                    im += Gm[(k + 4) * 16 + j] * vr[j];
                }
                o[k] = make_float2(re, im);
            }
            tile[tO] = o[0]; tile[tO + ol] = o[1]; tile[tO + oh] = o[2]; tile[tO + oh + ol] = o[3];
#endif
        }
        __syncthreads();
    }

    for (int it = 0; it < 16; ++it) {
        unsigned local = ((unsigned)(tid + it * 256)) * 2u;
        unsigned glob = (local & aa.lm) | (tileid << aa.tb) | ((local >> aa.ls) << aa.hs);
        *(float4*)&psi[glob] = *(float4*)&tile[local];
    }
}

// qubit q <-> bit (19 - q). offset-0 gate i: bits (19-2i, 18-2i). offset-1 gate i<9:
// bits (18-2i, 17-2i); gate 9: (hi=bit0, lo=bit19).
static void fill_off0_A(GateArgs& ga, int l) {
    ga.n = 6;
    for (int j = 0; j < 6; ++j) { int i = 4 + j; ga.hi[j] = 19 - 2 * i; ga.lo[j] = 18 - 2 * i; ga.gid[j] = l * 20 + i; }
}
static void fill_off0_B(GateArgs& ga, int l) {
    ga.n = 4;  // global bits 12..19 -> local 5..12 (local = global - 7)
    for (int i = 0; i < 4; ++i) { ga.hi[i] = 12 - 2 * i; ga.lo[i] = 11 - 2 * i; ga.gid[i] = l * 20 + i; }
}
static void fill_off1_A(GateArgs& ga, int l) {
    ga.n = 6;
    for (int j = 0; j < 6; ++j) { int i = 3 + j; ga.hi[j] = 18 - 2 * i; ga.lo[j] = 17 - 2 * i; ga.gid[j] = l * 20 + 10 + i; }
}
static void fill_off1_B(GateArgs& ga, int l) {
    ga.n = 4;  // global bits 13..19 -> local 6..12 (local = global - 7); bit0 -> local 0
    for (int i = 0; i < 3; ++i) { ga.hi[i] = 11 - 2 * i; ga.lo[i] = 10 - 2 * i; ga.gid[i] = l * 20 + 10 + i; }
    ga.hi[3] = 0; ga.lo[3] = 12; ga.gid[3] = l * 20 + 10 + 9;  // wrap gate (q19,q0)
}

extern "C" void kernel_launch(void* const* d_in, const int* in_sizes, int n_in,
                              void* d_out, int out_size, void* d_ws, size_t ws_size,
                              hipStream_t stream) {
    (void)in_sizes; (void)n_in; (void)out_size; (void)ws_size;
    const float2* state = (const float2*)d_in[0];
    const float2* H1 = (const float2*)d_in[1];
    const float2* H2 = (const float2*)d_in[2];
    float2* psi = (float2*)d_out;
    float* G = (float*)d_ws;  // 120 gates * 256 floats = 120 KB

    hipMemcpyAsync(psi, state, (size_t)(1u << 20) * sizeof(float2),
                   hipMemcpyDeviceToDevice, stream);
    expm_g_kernel<<<1, 128, 0, stream>>>(H1, H2, G);

    const AddrArgs aA  = {0x1FFFu, 13, 13, 13};  // free bits 0..12 (contiguous tile)
    const AddrArgs aB0 = {0x1Fu,   5,  5, 12};   // free bits 0..4 + 12..19
    const AddrArgs aB1 = {0x3Fu,   6,  6, 13};   // free bits 0..5 + 13..19

    for (int l = 0; l < 6; ++l) {
        GateArgs ga = {};
        fill_off0_A(ga, l);
        apply_block_kernel<<<128, 256, 0, stream>>>(psi, G, aA, ga);
        GateArgs gb = {};
        fill_off0_B(gb, l);
        apply_block_kernel<<<128, 256, 0, stream>>>(psi, G, aB0, gb);
        GateArgs gc = {};
        fill_off1_A(gc, l);
        apply_block_kernel<<<128, 256, 0, stream>>>(psi, G, aA, gc);
        GateArgs gd = {};
        fill_off1_B(gd, l);
        apply_block_kernel<<<128, 256, 0, stream>>>(psi, G, aB1, gd);
    }
}